// MultiHeadDotAttn_78606491452123
// MI455X (gfx1250) — compile-verified
//
#include <hip/hip_runtime.h>

// ---------------------------------------------------------------------------
// Multi-head single-step dot attention, algebraically refactored:
//   qk[b,h,k] = sum_a (sum_q query[b,q] Wq[h,a,q]) * Wk[h,a,k]
//   energy[b,h,s] = keys[b,s,:] . qk[b,h,:]
//   attn = masked softmax(energy)      (written in-place into d_out)
//   ctx[b,h,d] = sum_s attn[b,h,s] * value[b,s,d]
// All GEMM stages use v_wmma_f32_16x16x4_f32 (full f32 precision, CDNA5).
// ---------------------------------------------------------------------------

typedef __attribute__((ext_vector_type(2))) float v2f;
typedef __attribute__((ext_vector_type(8))) float v8f;

namespace {
constexpr int Bc  = 64;
constexpr int Sc  = 2048;
constexpr int QDc = 1024;
constexpr int KDc = 512;
constexpr int ADc = 512;
constexpr int Hc  = 8;
constexpr float NEGMASK = -1e10f;
}

static __device__ __forceinline__ v8f wmma_f32(v2f a, v2f b, v8f c) {
  // 8-arg form: (neg_a, A, neg_b, B, c_mod, C, reuse_a, reuse_b)
  return __builtin_amdgcn_wmma_f32_16x16x4_f32(false, a, false, b, (short)0, c,
                                               false, false);
}

// ---- Stage 1: q[b, h*AD + n] = query[b,:] . Wq[n,:]  (Wq flat (H*AD, QD)) --
__global__ __launch_bounds__(256) void k_proj_q(const float* __restrict__ query,
                                                const float* __restrict__ Wq,
                                                float* __restrict__ qout) {
  const int lane = threadIdx.x & 31;
  const int tile = blockIdx.x * 8 + (threadIdx.x >> 5);  // 0..1023
  const int mt = tile & 3;    // 4 tiles over B=64
  const int nt = tile >> 2;   // 256 tiles over H*AD=4096
  const int m  = mt * 16 + (lane & 15);
  const int n  = nt * 16 + (lane & 15);
  const int ko = (lane >> 4) * 2;  // K sub-offset per CDNA5 A/B frag layout
  const float* arow = query + (size_t)m * QDc;
  const float* brow = Wq + (size_t)n * QDc;
  v8f acc = {};
  for (int k = 0; k < QDc; k += 4) {
    v2f a = *(const v2f*)(arow + k + ko);
    v2f b = *(const v2f*)(brow + k + ko);
    acc = wmma_f32(a, b, acc);
  }
  const int mBase = mt * 16 + (lane >> 4) * 8;
#pragma unroll
  for (int v = 0; v < 8; ++v)
    qout[(size_t)(mBase + v) * (Hc * ADc) + n] = acc[v];
}

// ---- Stage 2: qk[b,h,n] = sum_a q[b,h,a] * Wk[h,a,n] ----------------------
__global__ __launch_bounds__(256) void k_proj_qk(const float* __restrict__ qin,
                                                 const float* __restrict__ Wk,
                                                 float* __restrict__ qk) {
  const int lane = threadIdx.x & 31;
  const int tile = blockIdx.x * 8 + (threadIdx.x >> 5);  // 0..1023
  const int h  = tile >> 7;   // 128 tiles per head
  const int r  = tile & 127;
  const int mt = r & 3;       // over B
  const int nt = r >> 2;      // over KD (32 tiles)
  const int m  = mt * 16 + (lane & 15);  // batch
  const int n  = nt * 16 + (lane & 15);  // kd column
  const int ko = (lane >> 4) * 2;
  const float* arow = qin + (size_t)m * (Hc * ADc) + (size_t)h * ADc;
  const float* bcol = Wk + (size_t)h * ADc * KDc + n;
  v8f acc = {};
  for (int a0 = 0; a0 < ADc; a0 += 4) {
    v2f a = *(const v2f*)(arow + a0 + ko);
    v2f b;
    b.x = bcol[(size_t)(a0 + ko) * KDc];
    b.y = bcol[(size_t)(a0 + ko + 1) * KDc];
    acc = wmma_f32(a, b, acc);
  }
  const int mBase = mt * 16 + (lane >> 4) * 8;
#pragma unroll
  for (int v = 0; v < 8; ++v)
    qk[(size_t)(mBase + v) * (Hc * KDc) + (size_t)h * KDc + n] = acc[v];
}

// ---- Stage 3: energy[b,h,s] = keys[b,s,:] . qk[b,h,:]  (streams keys once) -
// N dim = 16: heads 0..7 valid, 8..15 zero-padded branchlessly (EXEC stays
// all-ones across the WMMA as the ISA requires).
__global__ __launch_bounds__(256) void k_energy(const float* __restrict__ keys,
                                                const float* __restrict__ qk,
                                                float* __restrict__ energy) {
  const int lane = threadIdx.x & 31;
  const int tile = blockIdx.x * 8 + (threadIdx.x >> 5);  // 0..8191
  const int b  = tile >> 7;     // 128 s-tiles per batch
  const int mt = tile & 127;
  const int srow = mt * 16 + (lane & 15);
  const int n  = lane & 15;     // head column (pad >= 8)
  const int ko = (lane >> 4) * 2;
  const float fvalid = (n < Hc) ? 1.0f : 0.0f;
  const int   nc     = n & (Hc - 1);
  const float* arow = keys + ((size_t)b * Sc + srow) * KDc;
  const float* brow = qk + ((size_t)b * Hc + nc) * KDc;
  v8f acc = {};
  for (int k = 0; k < KDc; k += 4) {
    v2f a  = *(const v2f*)(arow + k + ko);
    v2f bb = *(const v2f*)(brow + k + ko);
    bb.x *= fvalid;
    bb.y *= fvalid;
    acc = wmma_f32(a, bb, acc);
  }
  if (n < Hc) {
    const int sBase = mt * 16 + (lane >> 4) * 8;
#pragma unroll
    for (int v = 0; v < 8; ++v)
      energy[((size_t)b * Hc + n) * Sc + sBase + v] = acc[v];
  }
}

// ---- Stage 4: masked softmax over S, in place in d_out --------------------
__global__ __launch_bounds__(256) void k_softmax(float* __restrict__ attn,
                                                 const int* __restrict__ key_len) {
  const int row = blockIdx.x;  // b*H + h
  const int b   = row >> 3;
  float* e = attn + (size_t)row * Sc;
  const int kl  = key_len[b];
  const int tid = threadIdx.x;
  __shared__ float sred[256];

  float mx = -3.402823e38f;
  for (int s = tid; s < Sc; s += 256) {
    float v = (s < kl) ? e[s] : NEGMASK;
    mx = fmaxf(mx, v);
  }
  sred[tid] = mx;
  __syncthreads();
  for (int off = 128; off > 0; off >>= 1) {
    if (tid < off) sred[tid] = fmaxf(sred[tid], sred[tid + off]);
    __syncthreads();
  }
  mx = sred[0];
  __syncthreads();

  float sum = 0.0f;
  for (int s = tid; s < Sc; s += 256) {
    float v = (s < kl) ? e[s] : NEGMASK;
    sum += __expf(v - mx);
  }
  sred[tid] = sum;
  __syncthreads();
  for (int off = 128; off > 0; off >>= 1) {
    if (tid < off) sred[tid] += sred[tid + off];
    __syncthreads();
  }
  const float inv = 1.0f / sred[0];
  __syncthreads();

  for (int s = tid; s < Sc; s += 256) {
    float v = (s < kl) ? e[s] : NEGMASK;
    e[s] = __expf(v - mx) * inv;
  }
}

// ---- Stage 5: ctx[b, h*KD + d] = sum_s attn[b,h,s] * value[b,s,d] ---------
// M=16 (heads, padded), N=16 d columns, K=S=2048; streams value once.
__global__ __launch_bounds__(256) void k_context(const float* __restrict__ attn,
                                                 const float* __restrict__ value,
                                                 float* __restrict__ ctx) {
  const int lane = threadIdx.x & 31;
  const int tile = blockIdx.x * 8 + (threadIdx.x >> 5);  // 0..2047
  const int b  = tile >> 5;     // 32 d-tiles per batch
  const int nt = tile & 31;
  const int m  = lane & 15;     // head row (pad >= 8)
  const int n  = nt * 16 + (lane & 15);  // d column
  const int ko = (lane >> 4) * 2;
  const float fvalid = (m < Hc) ? 1.0f : 0.0f;
  const int   mc     = m & (Hc - 1);
  const float* arow  = attn + ((size_t)b * Hc + mc) * Sc;
  const float* bbase = value + (size_t)b * Sc * KDc + n;
  v8f acc = {};
  for (int s = 0; s < Sc; s += 4) {
    v2f a = *(const v2f*)(arow + s + ko);
    a.x *= fvalid;
    a.y *= fvalid;
    v2f bb;
    bb.x = bbase[(size_t)(s + ko) * KDc];
    bb.y = bbase[(size_t)(s + ko + 1) * KDc];
    acc = wmma_f32(a, bb, acc);
  }
  // D layout: lanes 0-15 hold M=0..7 (valid heads) across v8f; lanes 16-31
  // hold M=8..15 which are padding.
  if (lane < 16) {
#pragma unroll
    for (int v = 0; v < 8; ++v)
      ctx[(size_t)b * (Hc * KDc) + (size_t)v * KDc + n] = acc[v];
  }
}

// ---------------------------------------------------------------------------
extern "C" void kernel_launch(void* const* d_in, const int* in_sizes, int n_in,
                              void* d_out, int out_size, void* d_ws,
                              size_t ws_size, hipStream_t stream) {
  const float* query   = (const float*)d_in[0];
  const float* keys    = (const float*)d_in[1];
  const float* value   = (const float*)d_in[2];
  const int*   key_len = (const int*)d_in[3];
  const float* Wq      = (const float*)d_in[4];
  const float* Wk      = (const float*)d_in[5];

  float* out      = (float*)d_out;
  float* ctx_out  = out;                                 // B * (H*KD)
  float* attn_out = out + (size_t)Bc * Hc * KDc;         // B * H * S

  float* q_ws  = (float*)d_ws;                           // B*H*AD  (1 MB)
  float* qk_ws = q_ws + (size_t)Bc * Hc * ADc;           // B*H*KD  (1 MB)

  k_proj_q <<<128, 256, 0, stream>>>(query, Wq, q_ws);
  k_proj_qk<<<128, 256, 0, stream>>>(q_ws, Wk, qk_ws);
  k_energy <<<1024, 256, 0, stream>>>(keys, qk_ws, attn_out);
  k_softmax<<<512, 256, 0, stream>>>(attn_out, key_len);
  k_context<<<256, 256, 0, stream>>>(attn_out, value, ctx_out);
}